// CapsLayer_55611236548958
// MI455X (gfx1250) — compile-verified
//
#include <hip/hip_runtime.h>

// Capsule routing, fused for MI455X (gfx1250, wave32, WMMA f32 16x16x4).
// B=256, IC=1152, ID=8, OC=10, OD=16.

typedef float v2f __attribute__((ext_vector_type(2)));
typedef float v8f __attribute__((ext_vector_type(8)));

#define B_TOT   256
#define IC      1152
#define ID      8
#define OC      10
#define OD      16
#define BTILE   16          // WMMA M = batch rows per workgroup
#define CHUNKS  9           // i-chunks per batch tile
#define WAVES   4
#define THREADS (WAVES * 32)
#define I_PER_WG   (IC / CHUNKS)        // 128
#define I_PER_WAVE (I_PER_WG / WAVES)   // 32
#define VS_ELEMS (B_TOT * OC * OD)      // 40960 floats (V or S buffer)

__global__ void caps_init(float* ws) {
    int t = blockIdx.x * 256 + threadIdx.x;
    if (t < 2 * VS_ELEMS) ws[t] = 0.0f;
}

// One workgroup: 16 batch rows x 128 input capsules. Computes partial
// s[b,o,d] = sum_i softmax_o(u*V) * u  and atomically accumulates to Sg.
__global__ __launch_bounds__(THREADS, 1)
void caps_route(const float* __restrict__ x, const float* __restrict__ W,
                const float* __restrict__ Vg, float* __restrict__ Sg) {
    __shared__ float Vl[OC * 16 * 16];   // [o][row(batch 0..15)][d]
    __shared__ float Sl[OC * 16 * 16];

    const int t     = threadIdx.x;
    const int btile = blockIdx.x / CHUNKS;
    const int chunk = blockIdx.x % CHUNKS;
    const int b0    = btile * BTILE;

    // Cooperative: load routing state V for this batch tile, zero S scratch.
    for (int j = t; j < OC * 256; j += THREADS) {
        int o = j >> 8, row = (j >> 4) & 15, d = j & 15;
        Vl[j] = Vg[(b0 + row) * (OC * OD) + o * OD + d];
        Sl[j] = 0.0f;
    }
    __syncthreads();

    const int lane = t & 31;
    const int wave = t >> 5;
    const int hi   = lane >> 4;     // half-wave select
    const int dL   = lane & 15;     // N column (= d) / A matrix row (= m)
    const int koff = hi * 2;        // A/B operand K offset per 16x16x4 layout

    v8f zero = {};
    v8f s_acc[OC];
#pragma unroll
    for (int o = 0; o < OC; ++o) s_acc[o] = zero;

    const int i_begin = chunk * I_PER_WG + wave * I_PER_WAVE;
    // A operand source: x[b0+m, i, k..k+1], m = dL, k = koff (+4 for 2nd step)
    const float* xlane = x + ((size_t)(b0 + dL) * IC + i_begin) * ID + koff;

    for (int ii = 0; ii < I_PER_WAVE; ++ii) {
        const int i = i_begin + ii;
        v2f a0 = *(const v2f*)(xlane + (size_t)ii * ID);       // k = koff, koff+1
        v2f a1 = *(const v2f*)(xlane + (size_t)ii * ID + 4);   // k = koff+4, koff+5

        // u tiles: one 16(batch) x 16(d) f32 accumulator per output capsule o
        v8f u[OC];
        const float* wl = W + (((size_t)i * OC) * OD + dL) * ID + koff;
#pragma unroll
        for (int o = 0; o < OC; ++o) {
            v2f bb0 = *(const v2f*)(wl + (size_t)o * OD * ID);      // B[k][n=dL]
            v2f bb1 = *(const v2f*)(wl + (size_t)o * OD * ID + 4);
            v8f acc = zero;
            acc = __builtin_amdgcn_wmma_f32_16x16x4_f32(
                      false, a0, false, bb0, (short)0, acc, false, false);
            acc = __builtin_amdgcn_wmma_f32_16x16x4_f32(
                      false, a1, false, bb1, (short)0, acc, false, false);
            u[o] = acc;
        }

        // Softmax over o per (batch row, d), elementwise on accumulator tiles.
        // |u*V| << 1 so max-subtraction is unnecessary (mathematically equal).
        float Z[8];
#pragma unroll
        for (int r = 0; r < 8; ++r) Z[r] = 0.0f;
#pragma unroll
        for (int o = 0; o < OC; ++o) {
#pragma unroll
            for (int r = 0; r < 8; ++r) {
                float vv = Vl[(o * 16 + r + 8 * hi) * 16 + dL];
                float e  = __expf(u[o][r] * vv);
                Z[r]    += e;
                u[o][r]  = e * u[o][r];   // e*u; divide by Z afterwards
            }
        }
#pragma unroll
        for (int r = 0; r < 8; ++r) Z[r] = 1.0f / Z[r];
#pragma unroll
        for (int o = 0; o < OC; ++o)
#pragma unroll
            for (int r = 0; r < 8; ++r)
                s_acc[o][r] += u[o][r] * Z[r];
    }

    // Reduce the 4 waves' partial s into LDS, then one atomic pass to global.
#pragma unroll
    for (int o = 0; o < OC; ++o)
#pragma unroll
        for (int r = 0; r < 8; ++r)
            atomicAdd(&Sl[(o * 16 + r + 8 * hi) * 16 + dL], s_acc[o][r]);
    __syncthreads();

    for (int j = t; j < OC * 256; j += THREADS) {
        int o = j >> 8, row = (j >> 4) & 15, d = j & 15;
        atomicAdd(&Sg[(b0 + row) * (OC * OD) + o * OD + d], Sl[j]);
    }
}

// Per-batch squash: v = s * l2 / ((1+l2)*sqrt(l2)); V += v; S = 0;
// last iteration writes v to output [B, OC, OD].
__global__ __launch_bounds__(160)
void caps_squash(float* __restrict__ Vg, float* __restrict__ Sg,
                 float* __restrict__ out, int last) {
    const int b   = blockIdx.x;
    const int t   = threadIdx.x;             // 0..159 = o*16 + d
    const int idx = b * (OC * OD) + t;
    float sv = Sg[idx];
    float sq = sv * sv;
    // sum over d: 16-lane half-wave reduction (wave32, width 16 segments)
    sq += __shfl_xor(sq, 8, 16);
    sq += __shfl_xor(sq, 4, 16);
    sq += __shfl_xor(sq, 2, 16);
    sq += __shfl_xor(sq, 1, 16);
    float scale = sq / ((1.0f + sq) * sqrtf(sq));
    float v = sv * scale;
    Vg[idx] += v;
    Sg[idx]  = 0.0f;
    if (last) out[idx] = v;
}

extern "C" void kernel_launch(void* const* d_in, const int* in_sizes, int n_in,
                              void* d_out, int out_size, void* d_ws, size_t ws_size,
                              hipStream_t stream) {
    (void)in_sizes; (void)n_in; (void)out_size; (void)ws_size;
    const float* x = (const float*)d_in[0];   // [256,1152,8]
    const float* W = (const float*)d_in[1];   // [1152,10,16,8]
    float* Vg  = (float*)d_ws;                // 40960 floats: accumulated v
    float* Sg  = Vg + VS_ELEMS;               // 40960 floats: s reduction buf
    float* out = (float*)d_out;               // [256,10,16]

    caps_init<<<(2 * VS_ELEMS + 255) / 256, 256, 0, stream>>>(Vg);
    for (int it = 0; it < 3; ++it) {
        caps_route<<<16 * CHUNKS, THREADS, 0, stream>>>(x, W, Vg, Sg);
        caps_squash<<<B_TOT, 160, 0, stream>>>(Vg, Sg, out, it == 2);
    }
}